// GraphAttentionLayer_11510512353761
// MI455X (gfx1250) — compile-verified
//
#include <hip/hip_runtime.h>
#include <hip/hip_bf16.h>

// ---------------------------------------------------------------------------
// GAT forward for MI455X (gfx1250, wave32).
//   h      = x @ W                          (8192x512 @ 512x512, fp32 WMMA)
//   e_src  = <h[b,n,head,:], a_src[head]>   e_dst likewise
//   logits = leakyrelu(e_src_i + e_dst_j), masked by adj, softmax over j
//   out    = attn @ h_head  (per b,head; probs staged in LDS, fp32 WMMA)
//   attn_mean = mean over heads (register-accumulated, no 134MB tensor)
// ---------------------------------------------------------------------------

typedef float v2f __attribute__((ext_vector_type(2)));
typedef float v8f __attribute__((ext_vector_type(8)));

#define DM   512          // d_model = H*dk
#define NN   512          // nodes
#define NB   16           // batch
#define BN   (NB * NN)    // 8192 rows of x / h
#define NHEAD 8
#define DK   64

// ---------------------------------------------------------------------------
// Kernel 1: h = x @ W.  One wave computes one 16x16 tile, K in steps of 4
// using V_WMMA_F32_16X16X4_F32.
// A 16x4 fp32 lane layout (ISA 7.12.2): lanes 0-15 -> M=lane, v0=K0,v1=K1;
//                                        lanes 16-31 -> M=lane-16, v0=K2,v1=K3.
// B 4x16 mirrored: lanes 0-15 -> N=lane, v0=K0 row, v1=K1 row; hi half K2/K3.
// ---------------------------------------------------------------------------
__global__ __launch_bounds__(256) void gat_gemm_xw(const float* __restrict__ X,
                                                   const float* __restrict__ W,
                                                   float* __restrict__ H) {
  const int lane = threadIdx.x & 31;
  const int wave = threadIdx.x >> 5;
  const int tile = blockIdx.x * 8 + wave;      // 16384 tiles total
  const int tn   = tile & 31;                  // 32 col tiles
  const int tm   = tile >> 5;                  // 512 row tiles
  const int m0 = tm * 16, n0 = tn * 16;
  const int lo = lane & 15, hi = lane >> 4;

  const float* Ap = X + (m0 + lo) * DM + 2 * hi;       // += k per step
  const float* Bp = W + (2 * hi) * DM + n0 + lo;       // += k*DM per step

  v8f c = {};
  for (int k = 0; k < DM; k += 4) {
    v2f a = *(const v2f*)(Ap + k);                     // 8B aligned (k+2hi even)
    v2f b;
    b.x = Bp[k * DM];
    b.y = Bp[k * DM + DM];
    c = __builtin_amdgcn_wmma_f32_16x16x4_f32(false, a, false, b,
                                              (short)0, c, false, false);
  }
#pragma unroll
  for (int v = 0; v < 8; ++v)
    H[(m0 + v + 8 * hi) * DM + n0 + lo] = c[v];
}

// ---------------------------------------------------------------------------
// Kernel 2: e_src[bn,h] = sum_d h[bn, h*64+d] * a[h, d]
//           e_dst[bn,h] = sum_d h[bn, h*64+d] * a[h, 64+d]
// ---------------------------------------------------------------------------
__global__ __launch_bounds__(256) void gat_edge_coef(const float* __restrict__ H,
                                                     const float* __restrict__ A,
                                                     float* __restrict__ Esrc,
                                                     float* __restrict__ Edst) {
  int idx = blockIdx.x * 256 + threadIdx.x;            // BN * NHEAD = 65536
  if (idx >= BN * NHEAD) return;
  int head = idx & (NHEAD - 1);
  int bn   = idx >> 3;
  const float* hrow = H + bn * DM + head * DK;
  const float* asrc = A + head * (2 * DK);
  const float* adst = asrc + DK;
  float s = 0.f, d = 0.f;
#pragma unroll 8
  for (int t = 0; t < DK; ++t) {
    float hv = hrow[t];
    s = fmaf(hv, asrc[t], s);
    d = fmaf(hv, adst[t], d);
  }
  Esrc[bn * NHEAD + head] = s;
  Edst[bn * NHEAD + head] = d;
}

// ---------------------------------------------------------------------------
// Kernel 3: fused masked-softmax + attn@h per (b, 16-row i-tile).
// 256 threads = 8 waves. Softmax: wave w owns local rows 2w, 2w+1; each lane
// owns 16 j's per row (j = lane + 32*m). Probabilities -> LDS (16x512 fp32).
// WMMA: waves 0-3 each produce one 16x16 tile of probs(16x512) @ h_head(512x16).
// attn_mean accumulated in registers across the 8 heads, written once.
// ---------------------------------------------------------------------------
__global__ __launch_bounds__(256) void gat_attn(const float* __restrict__ H,
                                                const float* __restrict__ Esrc,
                                                const float* __restrict__ Edst,
                                                const int*   __restrict__ Adj,
                                                float* __restrict__ Out,
                                                float* __restrict__ AttnMean) {
  __shared__ float probs[16 * NN];                     // 32 KB
  const int lane = threadIdx.x & 31;
  const int wave = threadIdx.x >> 5;
  const int b  = blockIdx.x >> 5;                      // 32 i-tiles per batch
  const int i0 = (blockIdx.x & 31) * 16;
  const int lo = lane & 15, hi = lane >> 4;

  float amean[32];
#pragma unroll
  for (int t = 0; t < 32; ++t) amean[t] = 0.f;

  for (int head = 0; head < NHEAD; ++head) {
    // ---------------- softmax over j for two rows per wave ----------------
#pragma unroll
    for (int rr = 0; rr < 2; ++rr) {
      const int il = 2 * wave + rr;
      const int i  = i0 + il;
      const float es = Esrc[(b * NN + i) * NHEAD + head];
      float lg[16];
      float mx = -3.0e38f;
#pragma unroll 4
      for (int m = 0; m < 16; ++m) {
        const int j = lane + 32 * m;
        float v = es + Edst[(b * NN + j) * NHEAD + head];
        v = (v >= 0.f) ? v : 0.2f * v;                 // leaky relu
        if (Adj[i * NN + j] == 0) v = -1.0e9f;         // mask
        lg[m] = v;
        mx = fmaxf(mx, v);
      }
#pragma unroll
      for (int off = 16; off > 0; off >>= 1)
        mx = fmaxf(mx, __shfl_xor(mx, off, 32));
      float sum = 0.f;
#pragma unroll 4
      for (int m = 0; m < 16; ++m) {
        float p = __expf(lg[m] - mx);
        lg[m] = p;
        sum += p;
      }
#pragma unroll
      for (int off = 16; off > 0; off >>= 1)
        sum += __shfl_xor(sum, off, 32);
      const float inv = 1.0f / sum;
#pragma unroll 4
      for (int m = 0; m < 16; ++m) {
        const float p = lg[m] * inv;
        probs[il * NN + lane + 32 * m] = p;
        amean[rr * 16 + m] += p;
      }
    }
    __syncthreads();

    // ---------------- attn @ h_head : 16x512 @ 512x16 per wave ------------
    if (wave < 4) {                                    // wave-uniform branch
      const int n0 = wave * 16;
      const float* Bp = H + (b * NN + 2 * hi) * DM + head * DK + n0 + lo;
      const float* Ap = &probs[lo * NN + 2 * hi];
      v8f c = {};
      for (int k = 0; k < NN; k += 4) {
        v2f a = *(const v2f*)(Ap + k);                 // LDS b64
        v2f bb;
        bb.x = Bp[k * DM];
        bb.y = Bp[k * DM + DM];
        c = __builtin_amdgcn_wmma_f32_16x16x4_f32(false, a, false, bb,
                                                  (short)0, c, false, false);
      }
#pragma unroll
      for (int v = 0; v < 8; ++v)
        Out[(b * NN + i0 + v + 8 * hi) * DM + head * DK + n0 + lo] = c[v];
    }
    __syncthreads();                                   // probs reused next head
  }

  // ---------------- attn_mean = sum_heads(p) / 8 --------------------------
#pragma unroll
  for (int rr = 0; rr < 2; ++rr) {
    const int i = i0 + 2 * wave + rr;
#pragma unroll
    for (int m = 0; m < 16; ++m) {
      const int j = lane + 32 * m;
      AttnMean[(b * NN + i) * NN + j] = amean[rr * 16 + m] * 0.125f;
    }
  }
}

// ---------------------------------------------------------------------------
extern "C" void kernel_launch(void* const* d_in, const int* in_sizes, int n_in,
                              void* d_out, int out_size, void* d_ws, size_t ws_size,
                              hipStream_t stream) {
  const float* x   = (const float*)d_in[0];   // (16,512,512)
  const int*   adj = (const int*)  d_in[1];   // (512,512)
  const float* W   = (const float*)d_in[2];   // (512,512)
  const float* a   = (const float*)d_in[3];   // (8,128)

  float* H    = (float*)d_ws;                 // 8192*512 f32 = 16 MB
  float* Esrc = H + (size_t)BN * DM;          // 65536 f32
  float* Edst = Esrc + (size_t)BN * NHEAD;    // 65536 f32

  float* Out      = (float*)d_out;            // 16*512*512
  float* AttnMean = Out + (size_t)NB * NN * DM;

  gat_gemm_xw  <<<(BN / 16) * (DM / 16) / 8, 256, 0, stream>>>(x, W, H);
  gat_edge_coef<<<(BN * NHEAD + 255) / 256, 256, 0, stream>>>(H, a, Esrc, Edst);
  gat_attn     <<<NB * (NN / 16),           256, 0, stream>>>(H, Esrc, Edst, adj,
                                                              Out, AttnMean);
}